// KVCacheManager_81724637708866
// MI455X (gfx1250) — compile-verified
//
#include <hip/hip_runtime.h>
#include <hip/hip_bf16.h>

// ---------------------------------------------------------------------------
// Problem constants (from the reference) — all geometry is static.
// ---------------------------------------------------------------------------
#define BATCH        32
#define SEQ_T        512
#define HEADS        8
#define HDIM         128
#define PAGE_SIZE    256
#define PAGES_PER_SEQ 8
#define NUM_PAGES    (BATCH * PAGES_PER_SEQ)            // 256
#define TOKEN_F      (HEADS * HDIM)                     // 1024 floats / token
#define TOKEN_V4     (TOKEN_F / 4)                      // 256 float4 / token
#define CACHE_ELEMS  ((long)NUM_PAGES * PAGE_SIZE * TOKEN_F)  // 67,108,864 floats
#define CACHE_V4     (CACHE_ELEMS / 4)                  // 16,777,216 float4

// Copy-kernel tiling: exact, no bounds checks needed.
#define CP_BLOCKS    4096
#define CP_THREADS   256
#define CP_CHUNKS    4
#define CP_STRIDE    ((long)CP_BLOCKS * CP_THREADS)               // 1,048,576
#define CP_PER_ITER  (CP_STRIDE * CP_CHUNKS)                      // 4,194,304
#define CP_ITERS     (CACHE_V4 / CP_PER_ITER)                     // 4
static_assert(CACHE_V4 % CP_PER_ITER == 0, "copy tiling must be exact");

typedef float v4f __attribute__((ext_vector_type(4)));
typedef int   v4i __attribute__((ext_vector_type(4)));

#define AS1 __attribute__((address_space(1)))
#define AS3 __attribute__((address_space(3)))

#if __has_builtin(__builtin_amdgcn_global_load_async_to_lds_b128) && \
    __has_builtin(__builtin_amdgcn_global_store_async_from_lds_b128) && \
    __has_builtin(__builtin_amdgcn_s_wait_asynccnt)
#define HAVE_GFX1250_ASYNC 1
#else
#define HAVE_GFX1250_ASYNC 0
#endif

// ---------------------------------------------------------------------------
// Bulk cache copy: src cache -> dst (output) cache.
// CDNA5 path: async global->LDS->global DMA (ASYNCcnt), 4 B128 transfers in
// flight per lane, perfectly coalesced (consecutive lanes -> consecutive 16B).
// Each thread owns 4 disjoint 16B LDS slots, so the only synchronization
// needed is the per-wave s_wait_asynccnt fence — no barriers.
// Exact static tiling: no bounds checks, constant trip count.
// ---------------------------------------------------------------------------
__global__ __launch_bounds__(CP_THREADS) void cache_copy_kernel(
    const v4f* __restrict__ src, v4f* __restrict__ dst) {
#if HAVE_GFX1250_ASYNC
  __shared__ v4f lbuf[CP_THREADS * CP_CHUNKS];  // 16 KB staging per block
  const int tid = threadIdx.x;
  const long linear = (long)blockIdx.x * CP_THREADS + tid;
#pragma unroll
  for (int it = 0; it < CP_ITERS; ++it) {
    const long base = (long)it * CP_PER_ITER + linear;
#pragma unroll
    for (int j = 0; j < CP_CHUNKS; ++j) {
      __builtin_amdgcn_global_load_async_to_lds_b128(
          (AS1 v4i*)(src + base + (long)j * CP_STRIDE),
          (AS3 v4i*)&lbuf[j * CP_THREADS + tid], /*offset=*/0, /*cpol=*/0);
    }
    __builtin_amdgcn_s_wait_asynccnt(0);  // loads landed in LDS
#pragma unroll
    for (int j = 0; j < CP_CHUNKS; ++j) {
      __builtin_amdgcn_global_store_async_from_lds_b128(
          (AS1 v4i*)(dst + base + (long)j * CP_STRIDE),
          (AS3 v4i*)&lbuf[j * CP_THREADS + tid], /*offset=*/0, /*cpol=*/0);
    }
    __builtin_amdgcn_s_wait_asynccnt(0);  // LDS slots reusable next iteration
  }
#else
  // Fallback: plain non-temporal 128-bit streaming copy, same exact tiling.
  const long linear = (long)blockIdx.x * CP_THREADS + threadIdx.x;
#pragma unroll
  for (int it = 0; it < CP_ITERS; ++it) {
#pragma unroll
    for (int j = 0; j < CP_CHUNKS; ++j) {
      const long idx = (long)it * CP_PER_ITER + (long)j * CP_STRIDE + linear;
      v4f t = __builtin_nontemporal_load(src + idx);
      __builtin_nontemporal_store(t, dst + idx);
    }
  }
#endif
}

// ---------------------------------------------------------------------------
// Scatter: one block per (b, i) token. 256 lanes x 16B = 4KB = one token row
// for each of k and v. Fully coalesced B128 accesses; page/slot math is
// uniform within the block. k/v are read exactly once -> non-temporal loads
// so the 128 MiB token stream doesn't evict the freshly written cache lines.
// ---------------------------------------------------------------------------
__global__ __launch_bounds__(TOKEN_V4) void kv_scatter_kernel(
    const v4f* __restrict__ k, const v4f* __restrict__ v,
    const int* __restrict__ page_table, const int* __restrict__ seqlens,
    v4f* __restrict__ out_k, v4f* __restrict__ out_v) {
  const int token = blockIdx.x;              // 0 .. B*T-1
  const int b = token >> 9;                  // / SEQ_T (512)
  const int i = token & (SEQ_T - 1);
  const int abs_pos = seqlens[b] + i;        // < 2048 by construction
  const int page = page_table[b * PAGES_PER_SEQ + (abs_pos >> 8)];  // / PAGE_SIZE
  const int slot = abs_pos & (PAGE_SIZE - 1);

  const long dst = ((long)page * PAGE_SIZE + (long)slot) * TOKEN_V4 + threadIdx.x;
  const long src = (long)token * TOKEN_V4 + threadIdx.x;

  out_k[dst] = __builtin_nontemporal_load(k + src);
  out_v[dst] = __builtin_nontemporal_load(v + src);
}

// ---------------------------------------------------------------------------
// new_seqlens = cache_seqlens + T (stored as float into the float32 d_out)
// ---------------------------------------------------------------------------
__global__ void seqlen_kernel(const int* __restrict__ seqlens, float* __restrict__ out) {
  const int b = threadIdx.x;
  if (b < BATCH) out[b] = (float)(seqlens[b] + SEQ_T);
}

extern "C" void kernel_launch(void* const* d_in, const int* in_sizes, int n_in,
                              void* d_out, int out_size, void* d_ws, size_t ws_size,
                              hipStream_t stream) {
  (void)in_sizes; (void)n_in; (void)out_size; (void)d_ws; (void)ws_size;

  const v4f* k       = (const v4f*)d_in[0];
  const v4f* v       = (const v4f*)d_in[1];
  const v4f* k_cache = (const v4f*)d_in[2];
  const v4f* v_cache = (const v4f*)d_in[3];
  const int* ptab    = (const int*)d_in[4];
  const int* slen    = (const int*)d_in[5];

  float* out    = (float*)d_out;
  v4f*   out_k  = (v4f*)out;
  v4f*   out_v  = (v4f*)(out + (size_t)CACHE_ELEMS);
  float* out_sl = out + 2 * (size_t)CACHE_ELEMS;

  // 1) Materialize the old caches into the outputs (async-DMA streaming copy).
  cache_copy_kernel<<<CP_BLOCKS, CP_THREADS, 0, stream>>>(k_cache, out_k);
  cache_copy_kernel<<<CP_BLOCKS, CP_THREADS, 0, stream>>>(v_cache, out_v);

  // 2) Scatter the B*T new tokens over the copies (same stream => ordered).
  kv_scatter_kernel<<<BATCH * SEQ_T, TOKEN_V4, 0, stream>>>(k, v, ptab, slen,
                                                            out_k, out_v);

  // 3) Updated sequence lengths.
  seqlen_kernel<<<1, 32, 0, stream>>>(slen, out_sl);
}